// A_Sp_P4CNN_69758858822525
// MI455X (gfx1250) — compile-verified
//
#include <hip/hip_runtime.h>
#include <math.h>

typedef __attribute__((ext_vector_type(2))) float v2f;
typedef __attribute__((ext_vector_type(8))) float v8f;

#define BATCH 32
#define CH    32
#define NGRP  4
#define EPSBN 2e-5f

// ---------- helpers ----------
__device__ __forceinline__ void rotidx(int r, int ky, int kx, int n1, int& sy, int& sx) {
    // rotated(ky,kx) = original(sy,sx) for jnp.rot90(w, k=r, axes=(-2,-1))
    switch (r & 3) {
        case 0:  sy = ky;      sx = kx;      break;
        case 1:  sy = kx;      sx = n1 - ky; break;
        case 2:  sy = n1 - ky; sx = n1 - kx; break;
        default: sy = n1 - kx; sx = ky;      break;
    }
}
__device__ __forceinline__ float sigm(float v) { return 1.0f / (1.0f + __expf(-v)); }

// ---------- layer 1: lifted spatial attention (Ci=1 -> mean=max=x) ----------
__global__ void k_att1(const float* __restrict__ x, const float* __restrict__ aw1,
                       float* __restrict__ att) {
    int idx = blockIdx.x * blockDim.x + threadIdx.x;
    const int total = BATCH * NGRP * 48 * 48;
    if (idx >= total) return;
    int xo = idx % 48, y = (idx / 48) % 48, r = (idx / 2304) % NGRP, b = idx / (2304 * NGRP);
    float acc = 0.f;
    for (int ky = 0; ky < 7; ++ky) {
        int yy = y + ky - 3; if (yy < 0 || yy >= 48) continue;
        for (int kx = 0; kx < 7; ++kx) {
            int xx = xo + kx - 3; if (xx < 0 || xx >= 48) continue;
            int sy, sx; rotidx(r, ky, kx, 6, sy, sx);
            float wv = aw1[sy * 7 + sx] + aw1[49 + sy * 7 + sx]; // mean-ch + max-ch
            acc += x[(b * 48 + yy) * 48 + xx] * wv;
        }
    }
    att[idx] = sigm(acc);
}

// ---------- layer 1: attentive lifting conv (K=9, VALU is fine) ----------
__global__ void k_conv1(const float* __restrict__ x, const float* __restrict__ att,
                        const float* __restrict__ w1, const float* __restrict__ b1,
                        float* __restrict__ out) {
    int idx = blockIdx.x * blockDim.x + threadIdx.x;
    const int total = BATCH * CH * NGRP * 46 * 46;
    if (idx >= total) return;
    int xo = idx % 46, y = (idx / 46) % 46;
    int r  = (idx / (46 * 46)) % NGRP;
    int co = (idx / (46 * 46 * NGRP)) % CH;
    int b  =  idx / (46 * 46 * NGRP * CH);
    float acc = b1[co];
    #pragma unroll
    for (int ky = 0; ky < 3; ++ky) {
        int yy = y + ky;
        #pragma unroll
        for (int kx = 0; kx < 3; ++kx) {
            int xx = xo + kx;
            int sy, sx; rotidx(r, ky, kx, 2, sy, sx);
            float m = att[((b * NGRP + r) * 48 + yy) * 48 + xx] * x[(b * 48 + yy) * 48 + xx];
            acc += m * w1[co * 9 + sy * 3 + sx];
        }
    }
    out[idx] = acc;
}

// ---------- channel mean/max reduction: [B,32,4,H,W] -> [2][B,4,H,W] ----------
__global__ void k_redmm(const float* __restrict__ X, float* __restrict__ red, int HW) {
    int idx = blockIdx.x * blockDim.x + threadIdx.x;
    int total = BATCH * NGRP * HW;
    if (idx >= total) return;
    int b = idx / (NGRP * HW); int rem = idx % (NGRP * HW);
    const float* p = X + (size_t)b * CH * NGRP * HW + rem;
    float s = 0.f, mx = -INFINITY;
    for (int c = 0; c < CH; ++c) { float v = p[(size_t)c * NGRP * HW]; s += v; mx = fmaxf(mx, v); }
    red[idx]         = s * (1.0f / CH);
    red[total + idx] = mx;
}

// ---------- p4 spatial attention (7x7 gg conv on 2x4 channels + sigmoid) ----------
__global__ void k_attgg(const float* __restrict__ red, const float* __restrict__ aw,
                        float* __restrict__ att, int H, int W) {
    int idx = blockIdx.x * blockDim.x + threadIdx.x;
    int HW = H * W;
    int total = BATCH * NGRP * HW;
    if (idx >= total) return;
    int xo = idx % W, y = (idx / W) % H, r = (idx / HW) % NGRP, b = idx / (NGRP * HW);
    float acc = 0.f;
    for (int c = 0; c < 2; ++c)
        for (int g = 0; g < NGRP; ++g) {
            int gsrc = (g - r) & 3;
            const float* rp = red + (size_t)c * BATCH * NGRP * HW + (size_t)(b * NGRP + g) * HW;
            const float* wp = aw + (c * NGRP + gsrc) * 49;
            for (int ky = 0; ky < 7; ++ky) {
                int yy = y + ky - 3; if (yy < 0 || yy >= H) continue;
                for (int kx = 0; kx < 7; ++kx) {
                    int xx = xo + kx - 3; if (xx < 0 || xx >= W) continue;
                    int sy, sx; rotidx(r, ky, kx, 6, sy, sx);
                    acc += rp[yy * W + xx] * wp[sy * 7 + sx];
                }
            }
        }
    att[idx] = sigm(acc);
}

// ---------- pre-rotate/roll GG weights into WR[r][co][tap][cg] ----------
__global__ void k_rotw(const float* __restrict__ w, float* __restrict__ WR, int Co, int CoPad) {
    int idx = blockIdx.x * blockDim.x + threadIdx.x;
    int total = NGRP * CoPad * 9 * 128;
    if (idx >= total) return;
    int cg  = idx & 127;
    int tap = (idx >> 7) % 9;
    int co  = (idx / (128 * 9)) % CoPad;
    int r   =  idx / (128 * 9 * CoPad);
    float v = 0.f;
    if (co < Co) {
        int ky = tap / 3, kx = tap % 3;
        int ci = cg >> 2, g = cg & 3, gsrc = (g - r) & 3;
        int sy, sx; rotidx(r, ky, kx, 2, sy, sx);
        v = w[(((co * CH + ci) * NGRP + gsrc) * 3 + sy) * 3 + sx];
    }
    WR[idx] = v;
}

// ---------- attentive GG conv as implicit GEMM on fp32 WMMA 16x16x4 ----------
// one wave = one 16(co) x 16(x) output tile at fixed (b, r, y).
// Geometry is compile-time so every B-operand load folds its channel stride
// (cg*HW*4 bytes, max ~1.05MB < 8MB IOFFSET range) into the load immediate.
template <int HI, int COPAD, int COSTORE>
__global__ void __launch_bounds__(32) k_wmma_conv(
    const float* __restrict__ X, const float* __restrict__ att,
    const float* __restrict__ WR, const float* __restrict__ bias,
    float* __restrict__ out)
{
    constexpr int WI = HI;
    constexpr int HO = HI - 2;
    constexpr int WO = HO;
    constexpr int HW = HI * WI;
    constexpr int MT = COPAD / 16;

    int lane = threadIdx.x;
    int nm = lane & 15;                 // A: m-row; B: n-col; D: n-col
    int kd = (lane >> 4) << 1;          // K-row base: lanes 0-15 -> {0,1}, 16-31 -> {2,3}
    int bz = blockIdx.z; int b = bz >> 2; int r = bz & 3;
    int y = blockIdx.y;
    int xt = (int)blockIdx.x / MT;
    int mt = (int)blockIdx.x - xt * MT;
    int xcol = xt * 16 + nm;
    float vmask = (xcol < WO) ? 1.0f : 0.0f;      // zero out B for padded N-columns
    int xc = (xcol < WO) ? xcol : (WO - 1);       // clamp address, keep EXEC all-ones
    const float* Xb   = X  + (size_t)b * (CH * NGRP) * HW + (size_t)kd * HW;
    const float* attb = att + (size_t)(b * NGRP + r) * HW + (size_t)y * WI;
    const float* Arow = WR + (size_t)(r * COPAD + mt * 16 + nm) * 1152 + kd;
    v8f acc = {};
    #pragma unroll
    for (int ky = 0; ky < 3; ++ky) {
        #pragma unroll
        for (int kx = 0; kx < 3; ++kx) {
            float av = attb[ky * WI + xc + kx] * vmask;   // attention scalar, per tap
            const float* Xt = Xb + ky * WI + xc + kx;     // + immediate channel offsets below
            const float* At = Arow + (ky * 3 + kx) * 128; // immediate offsets below
            #pragma unroll 16
            for (int cg = 0; cg < 128; cg += 4) {
                v2f a = *(const v2f*)(At + cg);           // aligned float2 of WR
                v2f bf;
                bf.x = av * Xt[cg * HW];                  // global_load offset: cg*HW*4
                bf.y = av * Xt[(cg + 1) * HW];
                acc = __builtin_amdgcn_wmma_f32_16x16x4_f32(
                    false, a, false, bf, (short)0, acc, false, false);
            }
        }
    }
    int hi = lane >> 4;
    if (xcol < WO) {
        #pragma unroll
        for (int j = 0; j < 8; ++j) {
            int co = mt * 16 + j + hi * 8;                // D layout: vgpr j -> M=j (+8 hi lanes)
            if (co < COSTORE)
                out[(((size_t)(b * COSTORE + co) * NGRP + r) * HO + y) * WO + xcol] =
                    acc[j] + bias[co];
        }
    }
}

// ---------- BatchNorm3d stats (training-mode, biased var) ----------
__global__ void k_bnstats(const float* __restrict__ X, float* __restrict__ st, int HW) {
    int c = blockIdx.x;
    int tid = threadIdx.x;
    int per = NGRP * HW;
    int N = BATCH * per;
    float s = 0.f, s2 = 0.f;
    for (int i = tid; i < N; i += blockDim.x) {
        int b = i / per; int rem = i - b * per;
        float v = X[(size_t)(b * CH + c) * per + rem];
        s += v; s2 += v * v;
    }
    __shared__ float sh[256], sh2[256];
    sh[tid] = s; sh2[tid] = s2;
    __syncthreads();
    for (int off = 128; off > 0; off >>= 1) {
        if (tid < off) { sh[tid] += sh[tid + off]; sh2[tid] += sh2[tid + off]; }
        __syncthreads();
    }
    if (tid == 0) {
        float m = sh[0] / (float)N;
        st[c] = m;
        st[CH + c] = sh2[0] / (float)N - m * m;
    }
}

__global__ void k_bnrelu(float* __restrict__ X, const float* __restrict__ st,
                         const float* __restrict__ ga, const float* __restrict__ be, int HW) {
    int idx = blockIdx.x * blockDim.x + threadIdx.x;
    int per = NGRP * HW;
    int total = BATCH * CH * per;
    if (idx >= total) return;
    int c = (idx / per) % CH;
    float y = ga[c] * (X[idx] - st[c]) * rsqrtf(st[CH + c] + EPSBN) + be[c];
    X[idx] = fmaxf(y, 0.f);
}

// ---------- 2x2/2 spatial max pool ----------
__global__ void k_pool(const float* __restrict__ in, float* __restrict__ out,
                       int Hi, int Wi, int Ho, int Wo) {
    int idx = blockIdx.x * blockDim.x + threadIdx.x;
    int total = BATCH * CH * NGRP * Ho * Wo;
    if (idx >= total) return;
    int xo = idx % Wo, y = (idx / Wo) % Ho;
    int rest = idx / (Ho * Wo);                     // (b*CH + c)*NGRP + r
    const float* p = in + ((size_t)rest * Hi + 2 * y) * Wi + 2 * xo;
    out[idx] = fmaxf(fmaxf(p[0], p[1]), fmaxf(p[Wi], p[Wi + 1]));
}

// ---------- rotation max + spatial mean -> [B,10] ----------
__global__ void k_final(const float* __restrict__ o7, float* __restrict__ out) {
    int idx = blockIdx.x * blockDim.x + threadIdx.x;
    if (idx >= BATCH * 10) return;
    const float* p = o7 + (size_t)idx * NGRP * 144;
    float acc = 0.f;
    for (int i = 0; i < 144; ++i) {
        float mx = fmaxf(fmaxf(p[i], p[144 + i]), fmaxf(p[288 + i], p[432 + i]));
        acc += mx;
    }
    out[idx] = acc * (1.0f / 144.0f);
}

// ================== host side ==================
extern "C" void kernel_launch(void* const* d_in, const int* in_sizes, int n_in,
                              void* d_out, int out_size, void* d_ws, size_t ws_size,
                              hipStream_t stream) {
    (void)in_sizes; (void)n_in; (void)out_size; (void)ws_size;
    const float* x   = (const float*)d_in[0];
    const float* w1  = (const float*)d_in[1];
    const float* b1  = (const float*)d_in[2];
    const float* aw1 = (const float*)d_in[3];
    const float* wl[8]; const float* bl[8]; const float* awl[8];
    for (int i = 2; i <= 7; ++i) {
        int base = 4 + (i - 2) * 3;
        wl[i]  = (const float*)d_in[base];
        bl[i]  = (const float*)d_in[base + 1];
        awl[i] = (const float*)d_in[base + 2];
    }
    const float* ga[7]; const float* be[7];
    for (int i = 1; i <= 6; ++i) {
        ga[i] = (const float*)d_in[22 + (i - 1) * 2];
        be[i] = (const float*)d_in[23 + (i - 1) * 2];
    }

    // workspace bump allocation (floats), with buffer reuse across layers
    float* ws   = (float*)d_ws;
    float* O1p  = ws;                   // 32*32*4*46*46 = 8,667,136
    float* O2p  = O1p + 8667136;        // 32*32*4*44*44 = 7,929,856
    float* P2p  = O2p + 7929856;        // 32*32*4*22*22 = 1,982,464
    float* ATTp = P2p + 1982464;        // max att: 32*4*48*48 = 294,912
    float* REDp = ATTp + 294912;        // max red: 2*32*4*46*46 = 541,696
    float* WRp  = REDp + 541696;        // 4*32*9*128 = 147,456
    float* STp  = WRp + 147456;         // 64 floats (mean|var)

    auto nb = [](int total) { return (total + 255) / 256; };

    // ---- layer 1 (lifting) ----
    k_att1 <<<nb(BATCH * NGRP * 48 * 48), 256, 0, stream>>>(x, aw1, ATTp);
    k_conv1<<<nb(BATCH * CH * NGRP * 46 * 46), 256, 0, stream>>>(x, ATTp, w1, b1, O1p);
    k_bnstats<<<CH, 256, 0, stream>>>(O1p, STp, 46 * 46);
    k_bnrelu<<<nb(BATCH * CH * NGRP * 46 * 46), 256, 0, stream>>>(O1p, STp, ga[1], be[1], 46 * 46);

    // attention + rotated-weight prep for a GG layer
    auto gg_prep = [&](const float* in, int Hi, int li, int Co, int CoPad) {
        int HW = Hi * Hi;
        k_redmm<<<nb(BATCH * NGRP * HW), 256, 0, stream>>>(in, REDp, HW);
        k_attgg<<<nb(BATCH * NGRP * HW), 256, 0, stream>>>(REDp, awl[li], ATTp, Hi, Hi);
        k_rotw <<<nb(NGRP * CoPad * 9 * 128), 256, 0, stream>>>(wl[li], WRp, Co, CoPad);
    };
    auto bn = [&](float* p, int Ho, int li) {
        k_bnstats<<<CH, 256, 0, stream>>>(p, STp, Ho * Ho);
        k_bnrelu<<<nb(BATCH * CH * NGRP * Ho * Ho), 256, 0, stream>>>(p, STp, ga[li], be[li], Ho * Ho);
    };
    auto grid_for = [](int Ho, int CoPad) {
        return dim3(((Ho + 15) / 16) * (CoPad / 16), Ho, BATCH * NGRP);
    };

    // ---- layer 2 + pool ----
    gg_prep(O1p, 46, 2, 32, 32);
    k_wmma_conv<46, 32, 32><<<grid_for(44, 32), 32, 0, stream>>>(O1p, ATTp, WRp, bl[2], O2p);
    bn(O2p, 44, 2);
    k_pool<<<nb(BATCH * CH * NGRP * 22 * 22), 256, 0, stream>>>(O2p, P2p, 44, 44, 22, 22);
    // ---- layer 3 ----
    gg_prep(P2p, 22, 3, 32, 32);
    k_wmma_conv<22, 32, 32><<<grid_for(20, 32), 32, 0, stream>>>(P2p, ATTp, WRp, bl[3], O1p);
    bn(O1p, 20, 3);
    // ---- layer 4 ----
    gg_prep(O1p, 20, 4, 32, 32);
    k_wmma_conv<20, 32, 32><<<grid_for(18, 32), 32, 0, stream>>>(O1p, ATTp, WRp, bl[4], O2p);
    bn(O2p, 18, 4);
    // ---- layer 5 ----
    gg_prep(O2p, 18, 5, 32, 32);
    k_wmma_conv<18, 32, 32><<<grid_for(16, 32), 32, 0, stream>>>(O2p, ATTp, WRp, bl[5], O1p);
    bn(O1p, 16, 5);
    // ---- layer 6 ----
    gg_prep(O1p, 16, 6, 32, 32);
    k_wmma_conv<16, 32, 32><<<grid_for(14, 32), 32, 0, stream>>>(O1p, ATTp, WRp, bl[6], O2p);
    bn(O2p, 14, 6);
    // ---- layer 7 (classifier, Co=10 padded to 16) ----
    gg_prep(O2p, 14, 7, 10, 16);
    k_wmma_conv<14, 16, 10><<<grid_for(12, 16), 32, 0, stream>>>(O2p, ATTp, WRp, bl[7], O1p);
    // ---- rotation-max + spatial mean ----
    k_final<<<nb(BATCH * 10), 256, 0, stream>>>(O1p, (float*)d_out);
}